// KuramotoDaidoMeanField_6906307412149
// MI455X (gfx1250) — compile-verified
//
#include <hip/hip_runtime.h>
#include <math.h>

// Kuramoto–Daido mean-field Euler integrator.
// Strictly sequential scalar recurrence -> single lane, latency-optimized.
//
// Per-step reformulation (algebraically identical to the reference):
//   b  = c0 - c1*s          where s = |Z|^2, c0 = 1 + DT*(K/2 - delta), c1 = DT*K/2
//   Z' = (b - i*wd)*Z       where wd = DT*omega
//   s' = (b^2 + wd^2)*s     (cross terms cancel exactly)
// Critical dependency chain: 3 VALU ops/step (fma, fma, mul); the complex
// update hangs off `b` and executes in the latency shadow of the s-chain.

__global__ __launch_bounds__(32, 1)
void KuramotoDaidoMeanField_kernel(const float* __restrict__ p_omega,
                                   const float* __restrict__ p_coupling,
                                   const float* __restrict__ p_delta,
                                   const float* __restrict__ p_zr,
                                   const float* __restrict__ p_zi,
                                   const int*   __restrict__ p_steps,
                                   float* __restrict__ out) {
    if (threadIdx.x != 0 || blockIdx.x != 0) return;

    const float DT    = 0.01f;
    const float w     = p_omega[0];
    const float K     = p_coupling[0];
    const float delta = p_delta[0];
    float zr          = p_zr[0];
    float zi          = p_zi[0];
    const int steps   = p_steps[0];

    const float Kh  = 0.5f * K;
    const float c0  = fmaf(DT, Kh - delta, 1.0f);  // 1 + DT*(K/2 - delta)
    const float nc1 = -(DT * Kh);                  // -DT*K/2
    const float wd  = DT * w;                      // DT*omega
    const float wd2 = wd * wd;

    float s = fmaf(zr, zr, zi * zi);               // |Z|^2

    int i = 0;
    for (; i + 4 <= steps; i += 4) {
#pragma unroll
        for (int u = 0; u < 4; ++u) {
            const float b   = fmaf(nc1, s, c0);    // chain op 1
            const float tr  = wd * zi;             // off-chain
            const float ti  = wd * zr;             // off-chain
            const float m   = fmaf(b, b, wd2);     // chain op 2
            const float nzr = fmaf(b, zr, tr);     // off s-chain
            const float nzi = fmaf(b, zi, -ti);    // off s-chain
            s  = m * s;                            // chain op 3
            zr = nzr;
            zi = nzi;
        }
    }
    for (; i < steps; ++i) {
        const float b   = fmaf(nc1, s, c0);
        const float tr  = wd * zi;
        const float ti  = wd * zr;
        const float m   = fmaf(b, b, wd2);
        const float nzr = fmaf(b, zr, tr);
        const float nzi = fmaf(b, zi, -ti);
        s  = m * s;
        zr = nzr;
        zi = nzi;
    }

    const float R   = sqrtf(fmaf(zr, zr, zi * zi));
    const float Psi = atan2f(zi, zr);

    out[0] = R;
    out[1] = Psi;
    out[2] = zr;
    out[3] = zi;
}

extern "C" void kernel_launch(void* const* d_in, const int* in_sizes, int n_in,
                              void* d_out, int out_size, void* d_ws, size_t ws_size,
                              hipStream_t stream) {
    (void)in_sizes; (void)n_in; (void)d_ws; (void)ws_size; (void)out_size;
    const float* p_omega    = (const float*)d_in[0];
    const float* p_coupling = (const float*)d_in[1];
    const float* p_delta    = (const float*)d_in[2];
    const float* p_zr       = (const float*)d_in[3];
    const float* p_zi       = (const float*)d_in[4];
    const int*   p_steps    = (const int*)d_in[5];
    float* out = (float*)d_out;

    KuramotoDaidoMeanField_kernel<<<1, 1, 0, stream>>>(
        p_omega, p_coupling, p_delta, p_zr, p_zi, p_steps, out);
}